// WaveletTransformer_867583394320
// MI455X (gfx1250) — compile-verified
//
#include <hip/hip_runtime.h>
#include <cstdint>

#define ROW_LEN   4096
#define BLOCK     256
#define INV_SQRT2 0.70710678118654752440f

// One workgroup processes one full row of 4096 f32 entirely in LDS:
//   global --(async b128)--> LDS A --(8 ping-pong Haar passes)--> LDS A --(async b128)--> global
// HBM traffic: exactly 1 read + 1 write per element (8 B/elem) == the bandwidth floor.
// Ping-pong buffers mean each pass only reads src / writes dst -> a single
// workgroup barrier per pass is sufficient (9 barriers total vs 18 in-place).

template <int HALF>
__device__ __forceinline__ void dwt_pass(const float* __restrict__ src,
                                         float* __restrict__ dst,
                                         unsigned tid)
{
    static_assert(HALF % BLOCK == 0, "all levels keep every thread active");
#pragma unroll
    for (int k = 0; k < HALF / BLOCK; ++k) {
        const unsigned i = tid + k * BLOCK;
        const float2 p = ((const float2*)src)[i];        // ds_load_b64, conflict-free
        dst[i]        = (p.x + p.y) * INV_SQRT2;         // cA
        dst[HALF + i] = (p.x - p.y) * INV_SQRT2;         // cD
    }
    __syncthreads();
}

template <int HALF>
__device__ __forceinline__ void idwt_pass(const float* __restrict__ src,
                                          float* __restrict__ dst,
                                          unsigned tid)
{
    static_assert(HALF % BLOCK == 0, "all levels keep every thread active");
#pragma unroll
    for (int k = 0; k < HALF / BLOCK; ++k) {
        const unsigned i = tid + k * BLOCK;
        const float a = src[i];                          // cA
        const float d = src[HALF + i];                   // cD
        ((float2*)dst)[i] =
            make_float2((a + d) * INV_SQRT2, (a - d) * INV_SQRT2);  // ds_store_b64
    }
    __syncthreads();
}

__global__ __launch_bounds__(BLOCK) void haar_roundtrip_kernel(const float* __restrict__ x,
                                                               float* __restrict__ out)
{
    __shared__ __align__(16) float bufA[ROW_LEN];        // 16 KB
    __shared__ __align__(16) float bufB[ROW_LEN];        // 16 KB

    const unsigned  tid     = threadIdx.x;
    const long long rowBase = (long long)blockIdx.x * ROW_LEN;

    // Low 32 bits of a generic pointer to a __shared__ object == raw LDS byte
    // offset (flat LDS addresses are {SHARED_BASE_hi32, lds_offset32}).
    const uint32_t ldsA = (uint32_t)(uintptr_t)bufA;

    // ---- async global -> LDS A: 4 x b128 per thread = 16 KB/row ----
#pragma unroll
    for (int k = 0; k < ROW_LEN / (BLOCK * 4); ++k) {
        const unsigned idx      = tid + k * BLOCK;       // float4 index
        const uint32_t lds_addr = ldsA + idx * 16u;
        const float*   gp       = x + rowBase + (size_t)idx * 4;
        asm volatile("global_load_async_to_lds_b128 %0, %1, off"
                     :: "v"(lds_addr), "v"(gp)
                     : "memory");
    }
    asm volatile("s_wait_asynccnt 0" ::: "memory");      // my wave's loads landed in LDS
    __syncthreads();                                     // all waves' loads visible

    // ---- forward DWT: A->B->A->B->A, details stay where written ----
    dwt_pass<2048>(bufA, bufB, tid);                     // cD0 -> B[2048:4096)
    dwt_pass<1024>(bufB, bufA, tid);                     // cD1 -> A[1024:2048)
    dwt_pass< 512>(bufA, bufB, tid);                     // cD2 -> B[ 512:1024)
    dwt_pass< 256>(bufB, bufA, tid);                     // cA4,cD3 -> A[0:512)

    // ---- inverse DWT: parity guarantees cA and the needed cD share a buffer ----
    idwt_pass< 256>(bufA, bufB, tid);                    // uses A[0:256)+A[256:512)
    idwt_pass< 512>(bufB, bufA, tid);                    // uses B[0:512)+B[512:1024)
    idwt_pass<1024>(bufA, bufB, tid);                    // uses A[0:1024)+A[1024:2048)
    idwt_pass<2048>(bufB, bufA, tid);                    // uses B[0:2048)+B[2048:4096)

    // ---- async LDS A -> global: 4 x b128 per thread ----
#pragma unroll
    for (int k = 0; k < ROW_LEN / (BLOCK * 4); ++k) {
        const unsigned idx      = tid + k * BLOCK;
        const uint32_t lds_addr = ldsA + idx * 16u;
        float*         gp       = out + rowBase + (size_t)idx * 4;
        asm volatile("global_store_async_from_lds_b128 %0, %1, off"
                     :: "v"(gp), "v"(lds_addr)
                     : "memory");
    }
    asm volatile("s_wait_asynccnt 0" ::: "memory");      // drain before s_endpgm
}

extern "C" void kernel_launch(void* const* d_in, const int* in_sizes, int n_in,
                              void* d_out, int out_size, void* d_ws, size_t ws_size,
                              hipStream_t stream)
{
    const float* x   = (const float*)d_in[0];
    float*       out = (float*)d_out;
    const int rows = in_sizes[0] / ROW_LEN;              // 32*512 = 16384 blocks
    hipLaunchKernelGGL(haar_roundtrip_kernel, dim3(rows), dim3(BLOCK), 0, stream, x, out);
}